// ClustGeoNodeEncoder_61555471286688
// MI455X (gfx1250) — compile-verified
//
#include <hip/hip_runtime.h>

typedef __attribute__((ext_vector_type(2))) float v2f;
typedef __attribute__((ext_vector_type(8))) float v8f;

// ---------------- init: zero counters / cursors / sc ----------------
__global__ void k_init(int* __restrict__ cnt, int* __restrict__ cur,
                       float* __restrict__ sc, int Cpad, int C) {
  int i = blockIdx.x * blockDim.x + threadIdx.x;
  if (i < Cpad) { cnt[i] = 0; cur[i] = 0; }
  if (i < C) sc[i] = 0.f;
}

// ---------------- histogram ----------------
__global__ void k_hist(const int* __restrict__ clusts, int* __restrict__ cnt, int n) {
  int i = blockIdx.x * blockDim.x + threadIdx.x;
  if (i < n) atomicAdd(&cnt[clusts[i]], 1);
}

// ---------------- exclusive scan (single workgroup, LDS Hillis-Steele) ----------------
__global__ __launch_bounds__(1024) void k_scan(const int* __restrict__ cnt,
                                               int* __restrict__ offs, int Cpad) {
  __shared__ int lds[1024];
  __shared__ int carry;
  int t = threadIdx.x;
  if (t == 0) carry = 0;
  __syncthreads();
  for (int base = 0; base < Cpad; base += 1024) {
    int i = base + t;
    int v = (i < Cpad) ? cnt[i] : 0;
    lds[t] = v;
    __syncthreads();
    int acc = v;
    for (int d = 1; d < 1024; d <<= 1) {
      int tv = (t >= d) ? lds[t - d] : 0;
      __syncthreads();
      acc += tv;
      lds[t] = acc;
      __syncthreads();
    }
    int myc = carry;
    if (i < Cpad) offs[i] = myc + acc - v;
    __syncthreads();
    if (t == 1023) carry = myc + acc;
    __syncthreads();
  }
  if (t == 0) offs[Cpad] = carry;
}

// ---------------- scatter into cluster-sorted order ----------------
__global__ void k_scatter(const float* __restrict__ data, const int* __restrict__ clusts,
                          const int* __restrict__ offs, int* __restrict__ cur,
                          float4* __restrict__ sorted, int n) {
  int i = blockIdx.x * blockDim.x + threadIdx.x;
  if (i >= n) return;
  int c = clusts[i];
  int pos = offs[c] + atomicAdd(&cur[c], 1);
  const float* d = data + (size_t)i * 6;
  sorted[pos] = make_float4(d[0], d[1], d[2], __int_as_float(c));
}

// Branchless component pick: idx 0->x, 1->y, 2->z, 3->1.0
__device__ __forceinline__ float compsel(const float4& v, int idx) {
  float r = 1.f;
  r = (idx == 0) ? v.x : r;
  r = (idx == 1) ? v.y : r;
  r = (idx == 2) ? v.z : r;
  return r;
}

// Branchless clamped load; out-of-range voxels get cid = -1 (one-hot row == 0)
__device__ __forceinline__ float4 ld_clamped(const float4* __restrict__ s, int idx, int end) {
  int i2 = (idx < end) ? idx : (end - 1);
  float4 v = s[i2];
  if (idx >= end) v.w = __int_as_float(-1);
  return v;
}

// ---------------- WMMA one-hot segment-sum of moments ----------------
// One wave per bin of 16 clusters. C[16 clusters x 16 feats] += OneHot[16xK] * Feat[Kx16]
// chained with V_WMMA_F32_16X16X4_F32 (exact f32 accumulation).
// Feature columns: [x, y, z, 1, xx, xy, xz, yy, yz, zz, 0...].
// ia/ib packed 2-bit selectors per feature column (0=x,1=y,2=z,3=one).
__global__ __launch_bounds__(32) void k_moments(const float4* __restrict__ sorted,
                                                const int* __restrict__ offs,
                                                float* __restrict__ mom, int C) {
  __shared__ float4 tile[2][32];
  int cbase = blockIdx.x * 16;
  int start = offs[cbase];
  int end   = offs[cbase + 16];
  int lane = threadIdx.x;
  int h = lane >> 4;      // half-wave selects K slots {2h, 2h+1} (A: K=0/2 in v0, 1/3 in v1)
  int n = lane & 15;      // M for A (one-hot cluster row), N for B (feature col)
  int cbn = cbase + n;
  int ia = (int)((0xFFF940E4u >> (2 * n)) & 3u);
  int ib = (int)((0xFFFA64FFu >> (2 * n)) & 3u);
  float zm = (n < 10) ? 1.f : 0.f;
  v8f acc = {};
  if (start < end) {
    float4 t = ld_clamped(sorted, start + lane, end);
    int buf = 0;
    for (int p = start; p < end; p += 32, buf ^= 1) {
      tile[buf][lane] = t;
      __syncthreads();
      if (p + 32 < end) t = ld_clamped(sorted, p + 32 + lane, end);  // prefetch next tile
#pragma unroll
      for (int j = 0; j < 8; ++j) {
        float4 va = tile[buf][4 * j + 2 * h];
        float4 vb = tile[buf][4 * j + 2 * h + 1];
        v2f A, B;
        A.x = (__float_as_int(va.w) == cbn) ? 1.f : 0.f;
        A.y = (__float_as_int(vb.w) == cbn) ? 1.f : 0.f;
        B.x = compsel(va, ia) * compsel(va, ib) * zm;
        B.y = compsel(vb, ia) * compsel(vb, ib) * zm;
        acc = __builtin_amdgcn_wmma_f32_16x16x4_f32(false, A, false, B, (short)0, acc,
                                                    false, false);
      }
      __syncthreads();
    }
  }
  // C layout: VGPR r -> M=r (lanes 0-15), M=r+8 (lanes 16-31); N = lane%16
#pragma unroll
  for (int r = 0; r < 8; ++r) {
    int c = cbase + r + 8 * h;
    if (n < 10 && c < C) mom[(size_t)c * 16 + n] = acc[r];
  }
}

// ---------------- per-cluster 3x3 symmetric eigensolve (analytic) ----------------
__global__ void k_eigen(const float* __restrict__ mom, float* __restrict__ out,
                        float* __restrict__ vdir, int C) {
  int c = blockIdx.x * blockDim.x + threadIdx.x;
  if (c >= C) return;
  const float* m = mom + (size_t)c * 16;
  float sx = m[0], sy = m[1], sz = m[2], nn = m[3];
  float inv = nn > 0.f ? 1.f / nn : 0.f;
  float cx = sx * inv, cy = sy * inv, cz = sz * inv;
  // A = sum x x^T - n c c^T  (n*c == s)
  float Axx = m[4] - sx * cx, Axy = m[5] - sx * cy, Axz = m[6] - sx * cz;
  float Ayy = m[7] - sy * cy, Ayz = m[8] - sy * cz, Azz = m[9] - sz * cz;
  // eigenvalues via trigonometric method
  float q = (Axx + Ayy + Azz) * (1.f / 3.f);
  float p1 = Axy * Axy + Axz * Axz + Ayz * Ayz;
  float dxx = Axx - q, dyy = Ayy - q, dzz = Azz - q;
  float p2 = dxx * dxx + dyy * dyy + dzz * dzz + 2.f * p1;
  float p = sqrtf(fmaxf(p2 * (1.f / 6.f), 0.f));
  float w1, w2;
  if (p > 1e-20f) {
    float ip = 1.f / p;
    float bxx = dxx * ip, byy = dyy * ip, bzz = dzz * ip;
    float bxy = Axy * ip, bxz = Axz * ip, byz = Ayz * ip;
    float detB = bxx * (byy * bzz - byz * byz) - bxy * (bxy * bzz - byz * bxz) +
                 bxz * (bxy * byz - byy * bxz);
    float r = fminf(fmaxf(0.5f * detB, -1.f), 1.f);
    float phi = acosf(r) * (1.f / 3.f);
    w2 = q + 2.f * p * cosf(phi);                              // largest
    float w0 = q + 2.f * p * cosf(phi + 2.0943951023931953f);  // smallest
    w1 = 3.f * q - w2 - w0;                                    // middle
  } else {
    w1 = q; w2 = q;
  }
  // eigenvector for w2: largest cross product of rows of (A - w2 I)
  float r0x = Axx - w2, r0y = Axy, r0z = Axz;
  float r1x = Axy, r1y = Ayy - w2, r1z = Ayz;
  float r2x = Axz, r2y = Ayz, r2z = Azz - w2;
  float c0x = r0y * r1z - r0z * r1y, c0y = r0z * r1x - r0x * r1z, c0z = r0x * r1y - r0y * r1x;
  float c1x = r0y * r2z - r0z * r2y, c1y = r0z * r2x - r0x * r2z, c1z = r0x * r2y - r0y * r2x;
  float c2x = r1y * r2z - r1z * r2y, c2y = r1z * r2x - r1x * r2z, c2z = r1x * r2y - r1y * r2x;
  float n0 = c0x * c0x + c0y * c0y + c0z * c0z;
  float n1 = c1x * c1x + c1y * c1y + c1z * c1z;
  float n2 = c2x * c2x + c2y * c2y + c2z * c2z;
  float vx = c0x, vy = c0y, vz = c0z, nb = n0;
  if (n1 > nb) { vx = c1x; vy = c1y; vz = c1z; nb = n1; }
  if (n2 > nb) { vx = c2x; vy = c2y; vz = c2z; nb = n2; }
  if (nb > 1e-30f) {
    float rn = rsqrtf(nb);
    vx *= rn; vy *= rn; vz *= rn;
  } else {
    vx = 0.f; vy = 0.f; vz = 1.f;
  }
  float iw2 = (fabsf(w2) > 1e-30f) ? 1.f / w2 : 0.f;
  float dirwt = (iw2 != 0.f) ? (1.f - w1 * iw2) : 0.f;
  float* o = out + (size_t)c * 16;
  o[0] = cx; o[1] = cy; o[2] = cz;
  o[3] = Axx * iw2; o[4]  = Axy * iw2; o[5]  = Axz * iw2;
  o[6] = Axy * iw2; o[7]  = Ayy * iw2; o[8]  = Ayz * iw2;
  o[9] = Axz * iw2; o[10] = Ayz * iw2; o[11] = Azz * iw2;
  o[15] = nn;
  float* vd = vdir + (size_t)c * 4;
  vd[0] = vx; vd[1] = vy; vd[2] = vz; vd[3] = dirwt;
}

// ---------------- pass 2: sc = segment_sum(x0 * |xp0|) ----------------
__global__ void k_pass2(const float4* __restrict__ sorted, const float* __restrict__ out,
                        const float* __restrict__ vdir, float* __restrict__ sc, int n) {
  int i = blockIdx.x * blockDim.x + threadIdx.x;
  if (i >= n) return;
  float4 s = sorted[i];
  int c = __float_as_int(s.w);
  const float* o = out + (size_t)c * 16;
  float xcx = s.x - o[0], xcy = s.y - o[1], xcz = s.z - o[2];
  const float* vd = vdir + (size_t)c * 4;
  float vx = vd[0], vy = vd[1], vz = vd[2];
  float x0 = xcx * vx + xcy * vy + xcz * vz;
  float px = xcx - x0 * vx, py = xcy - x0 * vy, pz = xcz - x0 * vz;
  float np = sqrtf(px * px + py * py + pz * pz);
  atomicAdd(&sc[c], x0 * np);
}

// ---------------- finalize v0 ----------------
__global__ void k_fin(const float* __restrict__ sc, const float* __restrict__ vdir,
                      float* __restrict__ out, int C) {
  int c = blockIdx.x * blockDim.x + threadIdx.x;
  if (c >= C) return;
  const float* vd = vdir + (size_t)c * 4;
  float f = (sc[c] < 0.f ? -1.f : 1.f) * vd[3];
  float* o = out + (size_t)c * 16;
  o[12] = vd[0] * f; o[13] = vd[1] * f; o[14] = vd[2] * f;
}

extern "C" void kernel_launch(void* const* d_in, const int* in_sizes, int n_in,
                              void* d_out, int out_size, void* d_ws, size_t ws_size,
                              hipStream_t stream) {
  const float* data  = (const float*)d_in[0];   // (N, 6) f32
  const int* clusts  = (const int*)d_in[1];     // (N,)   i32
  float* out = (float*)d_out;                   // (C, 16) f32
  int N = in_sizes[1];
  int C = out_size / 16;
  int Cpad = (C + 15) & ~15;

  char* ws = (char*)d_ws;
  size_t o = 0;
  auto take = [&](size_t bytes) { size_t r = o; o = (o + bytes + 15) & ~(size_t)15; return r; };
  int*    cnt    = (int*)(ws + take((size_t)Cpad * 4));
  int*    cur    = (int*)(ws + take((size_t)Cpad * 4));
  int*    offs   = (int*)(ws + take((size_t)(Cpad + 1) * 4));
  float*  sc     = (float*)(ws + take((size_t)C * 4));
  float*  vdir   = (float*)(ws + take((size_t)C * 16));
  float*  mom    = (float*)(ws + take((size_t)C * 64));
  float4* sorted = (float4*)(ws + take((size_t)N * 16));
  (void)ws_size; (void)n_in;

  int gC = (Cpad + 255) / 256;
  int gN = (N + 255) / 256;
  k_init   <<<gC, 256, 0, stream>>>(cnt, cur, sc, Cpad, C);
  k_hist   <<<gN, 256, 0, stream>>>(clusts, cnt, N);
  k_scan   <<<1, 1024, 0, stream>>>(cnt, offs, Cpad);
  k_scatter<<<gN, 256, 0, stream>>>(data, clusts, offs, cur, sorted, N);
  k_moments<<<Cpad / 16, 32, 0, stream>>>(sorted, offs, mom, C);
  k_eigen  <<<(C + 255) / 256, 256, 0, stream>>>(mom, out, vdir, C);
  k_pass2  <<<gN, 256, 0, stream>>>(sorted, out, vdir, sc, N);
  k_fin    <<<(C + 255) / 256, 256, 0, stream>>>(sc, vdir, out, C);
}